// GCN_90383291777260
// MI455X (gfx1250) — compile-verified
//
#include <hip/hip_runtime.h>
#include <hip/hip_bf16.h>

typedef __attribute__((ext_vector_type(2))) float v2f;
typedef __attribute__((ext_vector_type(8))) float v8f;

#define EMB 64
#define KPAD 12   // N_FEAT=9 zero-padded to a multiple of 4

// ---------------------------------------------------------------- utilities
__global__ void fill_f32(float* __restrict__ p, float v, int n) {
    int i = blockIdx.x * blockDim.x + threadIdx.x;
    if (i < n) p[i] = v;
}

// xp[N][12] = x[N][9] zero-padded in K
__global__ void pad_x_k(const float* __restrict__ x, float* __restrict__ xp, int N) {
    int tid = blockIdx.x * blockDim.x + threadIdx.x;
    if (tid >= N * KPAD) return;
    int n = tid / KPAD, c = tid - n * KPAD;
    xp[tid] = (c < 9) ? x[n * 9 + c] : 0.0f;
}

__global__ void deg_accum_k(const int* __restrict__ dst, float* __restrict__ deg, int E) {
    int e = blockIdx.x * blockDim.x + threadIdx.x;
    if (e < E) atomicAdd(&deg[dst[e]], 1.0f);
}

__global__ void node_norm_k(const float* __restrict__ deg, float* __restrict__ dinv,
                            float* __restrict__ selfn, int N) {
    int i = blockIdx.x * blockDim.x + threadIdx.x;
    if (i < N) {
        float d = deg[i];
        dinv[i]  = rsqrtf(d);
        selfn[i] = 1.0f / d;
    }
}

__global__ void edge_norm_k(const int* __restrict__ src, const int* __restrict__ dst,
                            const float* __restrict__ dinv, float* __restrict__ en, int E) {
    int e = blockIdx.x * blockDim.x + threadIdx.x;
    if (e < E) en[e] = dinv[src[e]] * dinv[dst[e]];
}

// ------------------------------------------------- dense transform via WMMA
// H[M x 64] = A[M x K] @ W[K x 64], fp32, v_wmma_f32_16x16x4_f32.
// K compile-time (12 or 64) -> fully unrolled, guard-free, branchless.
// A frag (16x4 f32): lanes 0-15 row m; VGPR0 = K=k0|k0+2, VGPR1 = k0+1|k0+3
//   -> per lane the two values are contiguous: one b64 load.
// C/D frag (16x16 f32): VGPR v: lanes 0-15 -> M=v, lanes 16-31 -> M=v+8, N=lane%16.
template <int K>
__global__ __launch_bounds__(128) void gemm_wmma_k(const float* __restrict__ A,
                                                   const float* __restrict__ W,
                                                   float* __restrict__ H) {
    const int lane = threadIdx.x & 31;
    const int wv   = threadIdx.x >> 5;     // N-tile 0..3
    const int row0 = blockIdx.x * 16;      // gridDim.x = M/16 (50000/16 = 3125 exact)
    const int col0 = wv * 16;
    const int mn   = lane & 15;            // A row / B,C col
    const int half = lane >> 4;

    const float* __restrict__ arow = A + (size_t)(row0 + mn) * K + half * 2;
    const float* __restrict__ wcol = W + (size_t)(half * 2) * EMB + col0 + mn;

    v8f c = {};
#pragma unroll
    for (int k0 = 0; k0 < K; k0 += 4) {
        v2f a = *(const v2f*)(arow + k0);          // global_load_b64
        v2f b;
        b.x = wcol[(size_t)k0 * EMB];              // W[k0+2h  ][col]
        b.y = wcol[(size_t)k0 * EMB + EMB];        // W[k0+2h+1][col]
        c = __builtin_amdgcn_wmma_f32_16x16x4_f32(false, a, false, b,
                                                  (short)0, c, false, false);
    }
#pragma unroll
    for (int v = 0; v < 8; ++v)
        H[(size_t)(row0 + v + half * 8) * EMB + col0 + mn] = c[v];
}

// ---------------------------------------------------------- sparse scatter
// agg[dst[e]][f] += h[src[e]][f] * edge_norm[e]
// one thread per (edge, 4-feature chunk): b128 gather + 4 f32 atomics (L2-resident)
__global__ void scatter_k(const float4* __restrict__ h4, const int* __restrict__ src,
                          const int* __restrict__ dst, const float* __restrict__ en,
                          float* __restrict__ agg, int E) {
    long long tid = (long long)blockIdx.x * blockDim.x + threadIdx.x;
    if (tid >= (long long)E * 16) return;
    int e = (int)(tid >> 4);
    int c = (int)(tid & 15);
    float  w = en[e];
    float4 v = h4[(size_t)src[e] * 16 + c];
    float* base = agg + (size_t)dst[e] * EMB + c * 4;
    atomicAdd(base + 0, v.x * w);
    atomicAdd(base + 1, v.y * w);
    atomicAdd(base + 2, v.z * w);
    atomicAdd(base + 3, v.w * w);
}

// out = tanh(agg + h*self_norm + b)
__global__ void combine_k(const float* __restrict__ h, const float* __restrict__ agg,
                          const float* __restrict__ selfn, const float* __restrict__ b,
                          float* __restrict__ out, int N) {
    long long tid = (long long)blockIdx.x * blockDim.x + threadIdx.x;
    if (tid >= (long long)N * EMB) return;
    int i = (int)(tid >> 6);
    int f = (int)(tid & (EMB - 1));
    out[tid] = tanhf(agg[tid] + h[tid] * selfn[i] + b[f]);
}

// ------------------------------------------------------------- pooling
__device__ __forceinline__ void atomicMaxFloat(float* addr, float value) {
    // sign-split trick: monotone over int for >=0, over uint (min) for <0.
    if (value >= 0.0f)
        atomicMax((int*)addr, __float_as_int(value));
    else
        atomicMin((unsigned int*)addr, (unsigned int)__float_as_int(value));
}

__global__ void pool_accum_k(const float4* __restrict__ h4, const int* __restrict__ batch,
                             float* __restrict__ gmp, float* __restrict__ gap,
                             float* __restrict__ cnt, int N) {
    long long tid = (long long)blockIdx.x * blockDim.x + threadIdx.x;
    if (tid >= (long long)N * 16) return;
    int i = (int)(tid >> 4);
    int c = (int)(tid & 15);
    int g = batch[i];
    float4 v = h4[tid];
    float* mp = gmp + (size_t)g * EMB + c * 4;
    float* ap = gap + (size_t)g * EMB + c * 4;
    atomicMaxFloat(mp + 0, v.x); atomicAdd(ap + 0, v.x);
    atomicMaxFloat(mp + 1, v.y); atomicAdd(ap + 1, v.y);
    atomicMaxFloat(mp + 2, v.z); atomicAdd(ap + 2, v.z);
    atomicMaxFloat(mp + 3, v.w); atomicAdd(ap + 3, v.w);
    if (c == 0) atomicAdd(&cnt[g], 1.0f);
}

__global__ void head_k(const float* __restrict__ gmp, const float* __restrict__ gap,
                       const float* __restrict__ cnt, const float* __restrict__ Wout,
                       const float* __restrict__ bout, float* __restrict__ out, int G) {
    int g = blockIdx.x * blockDim.x + threadIdx.x;
    if (g >= G) return;
    float acc  = bout[0];
    float invc = 1.0f / cnt[g];
#pragma unroll
    for (int f = 0; f < EMB; ++f) {
        acc += gmp[(size_t)g * EMB + f] * Wout[f];
        acc += gap[(size_t)g * EMB + f] * invc * Wout[EMB + f];
    }
    out[g] = acc;
}

// ---------------------------------------------------------------- driver
static inline int cdiv(long long a, int b) { return (int)((a + b - 1) / b); }

extern "C" void kernel_launch(void* const* d_in, const int* in_sizes, int n_in,
                              void* d_out, int out_size, void* d_ws, size_t ws_size,
                              hipStream_t stream) {
    const float* x     = (const float*)d_in[0];
    const int*   ei    = (const int*)  d_in[1];
    const int*   batch = (const int*)  d_in[2];
    const float* W[4]  = { (const float*)d_in[3], (const float*)d_in[5],
                           (const float*)d_in[7], (const float*)d_in[9] };
    const float* B[4]  = { (const float*)d_in[4], (const float*)d_in[6],
                           (const float*)d_in[8], (const float*)d_in[10] };
    const float* Wout  = (const float*)d_in[11];
    const float* bout  = (const float*)d_in[12];
    float* out = (float*)d_out;

    const int N = in_sizes[0] / 9;        // 50000
    const int E = in_sizes[1] / 2;        // 1200000
    const int G = out_size;               // 1024
    const int* src = ei;
    const int* dst = ei + E;

    // carve workspace (256B aligned slabs)
    char* ws = (char*)d_ws;
    auto carve = [&](size_t bytes) {
        void* p = (void*)ws;
        ws += (bytes + 255) & ~(size_t)255;
        return (float*)p;
    };
    float* deg   = carve((size_t)N * 4);
    float* dinv  = carve((size_t)N * 4);
    float* selfn = carve((size_t)N * 4);
    float* xp    = carve((size_t)N * KPAD * 4);
    float* en    = carve((size_t)E * 4);
    float* buf0  = carve((size_t)N * EMB * 4);
    float* buf1  = carve((size_t)N * EMB * 4);
    float* buf2  = carve((size_t)N * EMB * 4);
    float* gmp   = carve((size_t)G * EMB * 4);
    float* gap   = carve((size_t)G * EMB * 4);
    float* cnt   = carve((size_t)G * 4);
    (void)ws_size; (void)n_in;

    const int T = 256;
    const long long NE = (long long)N * EMB;

    // degree (with self loop) and normalization factors
    fill_f32<<<cdiv(N, T), T, 0, stream>>>(deg, 1.0f, N);
    deg_accum_k<<<cdiv(E, T), T, 0, stream>>>(dst, deg, E);
    node_norm_k<<<cdiv(N, T), T, 0, stream>>>(deg, dinv, selfn, N);
    edge_norm_k<<<cdiv(E, T), T, 0, stream>>>(src, dst, dinv, en, E);
    pad_x_k<<<cdiv((long long)N * KPAD, T), T, 0, stream>>>(x, xp, N);

    // 4 GCN layers; rotate {in, h, agg} over buf0/1/2 (combine writes agg in place)
    const float* layer_in = xp;
    float* h     = buf0;
    float* agg   = buf1;
    float* spare = buf2;
    for (int L = 0; L < 4; ++L) {
        if (L == 0) gemm_wmma_k<KPAD><<<N / 16, 128, 0, stream>>>(layer_in, W[L], h);
        else        gemm_wmma_k<EMB> <<<N / 16, 128, 0, stream>>>(layer_in, W[L], h);
        fill_f32<<<cdiv(NE, T), T, 0, stream>>>(agg, 0.0f, (int)NE);
        scatter_k<<<cdiv((long long)E * 16, T), T, 0, stream>>>(
            (const float4*)h, src, dst, en, agg, E);
        combine_k<<<cdiv(NE, T), T, 0, stream>>>(h, agg, selfn, B[L], agg, N);
        // rotate: agg now holds this layer's activations
        float* next_h   = (L == 0) ? spare : (float*)layer_in;
        float* next_in  = agg;
        float* next_agg = h;
        layer_in = next_in;
        h   = next_h;
        agg = next_agg;
    }

    // pooling
    fill_f32<<<cdiv(G * EMB, T), T, 0, stream>>>(gmp, -INFINITY, G * EMB);
    fill_f32<<<cdiv(G * EMB, T), T, 0, stream>>>(gap, 0.0f, G * EMB);
    fill_f32<<<cdiv(G, T), T, 0, stream>>>(cnt, 0.0f, G);
    pool_accum_k<<<cdiv((long long)N * 16, T), T, 0, stream>>>(
        (const float4*)layer_in, batch, gmp, gap, cnt, N);

    // output head
    head_k<<<cdiv(G, T), T, 0, stream>>>(gmp, gap, cnt, Wout, bout, out, G);
}